// VQGanVAEEmbed_72095321031182
// MI455X (gfx1250) — compile-verified
//
#include <hip/hip_runtime.h>
#include <hip/hip_bf16.h>
#include <stdint.h>

// out[b, d, n] = embedding[seq[b, n], d]
//   B=16, N=1024 (32x32), D=256, K=16384 tokens.
// Memory-bound gather+transpose (~32MB traffic): stage 32 rows x 1KB into LDS
// with gfx1250 async global->LDS b128 DMA, then transpose out of LDS with
// 128B-coalesced float4 stores.

#define TILE_N     32
#define EMB_D      256
#define LDS_STRIDE 260   // floats: 1040B row pitch, 16B-aligned, ~conflict-free transpose reads

__global__ __launch_bounds__(256) void embed_gather_nchw(
    const int* __restrict__ seq,     // [B, N] int32 token ids
    const float* __restrict__ emb,   // [K, D] fp32
    float* __restrict__ out,         // [B, D, N] fp32 (NCHW, N = H*W)
    int N)
{
  __shared__ int s_tok[TILE_N];
  __shared__ __align__(16) float s_tile[TILE_N * LDS_STRIDE];

  const int tid  = threadIdx.x;
  const int tile = blockIdx.x & 31;   // 32 n-tiles per batch image
  const int b    = blockIdx.x >> 5;
  const int n0   = tile * TILE_N;

  // Load this tile's 32 token ids once.
  if (tid < TILE_N) s_tok[tid] = seq[b * N + n0 + tid];
  __syncthreads();

  // Phase 1: async-DMA gather of 32 embedding rows (1KB each) into LDS.
  // 8 iters x 256 lanes x 16B = 32KB; each row read as 64 coalesced float4.
  #pragma unroll
  for (int i = 0; i < 8; ++i) {
    int e   = i * 256 + tid;     // float4 index within the tile
    int row = e >> 6;            // 64 float4 per embedding row
    int col = (e & 63) << 2;     // float offset within row
    const float* src = emb + (size_t)s_tok[row] * EMB_D + col;
    unsigned lds_off = (unsigned)(uintptr_t)(&s_tile[row * LDS_STRIDE + col]);
    asm volatile("global_load_async_to_lds_b128 %0, %1, off"
                 :: "v"(lds_off),
                    "v"((unsigned long long)(uintptr_t)src)
                 : "memory");
  }
  // Each wave waits for its own async transfers, then barrier -> tile visible.
  asm volatile("s_wait_asynccnt 0" ::: "memory");
  __syncthreads();

  // Phase 2: transposed write-out. out[b, d, n0 + n], contiguous along n.
  // tid -> (drow = tid/8, q = tid%8): 8 lanes cover one d-row's 32 n values
  // as float4 (128B contiguous per 8-lane group).
  const int q    = tid & 7;
  const int drow = tid >> 3;
  float* outb = out + (size_t)b * EMB_D * (size_t)N + n0;
  #pragma unroll
  for (int j = 0; j < 8; ++j) {
    int d = j * 32 + drow;
    float4 v;
    v.x = s_tile[(q * 4 + 0) * LDS_STRIDE + d];
    v.y = s_tile[(q * 4 + 1) * LDS_STRIDE + d];
    v.z = s_tile[(q * 4 + 2) * LDS_STRIDE + d];
    v.w = s_tile[(q * 4 + 3) * LDS_STRIDE + d];
    *(float4*)(outb + (size_t)d * N + q * 4) = v;
  }
}

extern "C" void kernel_launch(void* const* d_in, const int* in_sizes, int n_in,
                              void* d_out, int out_size, void* d_ws, size_t ws_size,
                              hipStream_t stream) {
  const int*   seq = (const int*)d_in[0];    // [16, 1024] int32
  const float* emb = (const float*)d_in[1];  // [16384, 256] fp32
  float*       out = (float*)d_out;          // [16, 256, 32, 32] fp32

  const int B = 16;
  const int N = 1024;                        // 32x32 latent grid
  dim3 grid(B * (N / TILE_N));               // 512 blocks
  dim3 block(256);                           // 8 wave32 per block
  embed_gather_nchw<<<grid, block, 0, stream>>>(seq, emb, out, N);
}